// LogicLayer_70961449665053
// MI455X (gfx1250) — compile-verified
//
#include <hip/hip_runtime.h>
#include <stdint.h>

#define IN_DIM  32768
#define OUT_DIM 32768
#define BATCH   1024
#define TPB     256

typedef float v4f  __attribute__((ext_vector_type(4)));
typedef unsigned int v4u __attribute__((ext_vector_type(4)));

// ---------------------------------------------------------------------------
// Pass 1: pack idx_a (15b) | idx_b<<15 (15b) | flagA<<30 | flagB<<31 into u32.
// sigmoid(z) > 0.5  <=>  z > 0, so we never need expf.
// Table is 128KB total -> stays hot in L2 for all 1024 row-blocks.
// ---------------------------------------------------------------------------
__global__ __launch_bounds__(TPB) void pack_idx_kernel(
    const float* __restrict__ logits,   // (OUT_DIM, 2)
    const int*   __restrict__ idx_a,
    const int*   __restrict__ idx_b,
    uint32_t*    __restrict__ packed)
{
    int j = blockIdx.x * TPB + threadIdx.x;
    if (j < OUT_DIM) {
        uint32_t ia = ((uint32_t)idx_a[j]) & 0x7FFFu;
        uint32_t ib = ((uint32_t)idx_b[j]) & 0x7FFFu;
        uint32_t fa = (logits[2 * j]     > 0.0f) ? 1u : 0u;
        uint32_t fb = (logits[2 * j + 1] > 0.0f) ? 1u : 0u;
        packed[j] = ia | (ib << 15) | (fa << 30) | (fb << 31);
    }
}

__device__ __forceinline__ float eval_pair(const float* __restrict__ smem, uint32_t p)
{
    float a = smem[p & 0x7FFFu];          // ds_load_b32 gathers
    float b = smem[(p >> 15) & 0x7FFFu];
    a = (p & 0x40000000u) ? 1.0f - a : a;
    b = (p & 0x80000000u) ? 1.0f - b : b;
    return a * b;
}

// ---------------------------------------------------------------------------
// Pass 2: one block per batch row. Async-copy the whole 128KB x-row into LDS
// (global_load_async_to_lds_b128, ASYNCcnt-tracked, VGPR-bypassing). The row
// is single-use -> non-temporal load hint so x's 128MB stream doesn't evict
// the reused packed table from L2. Then gather from LDS and stream results
// out with non-temporal 16B stores.
// ---------------------------------------------------------------------------
__global__ __launch_bounds__(TPB) void logic_layer_kernel(
    const float*    __restrict__ x,       // (BATCH, IN_DIM)
    const uint32_t* __restrict__ packed,  // (OUT_DIM)
    float*          __restrict__ out)     // (BATCH, OUT_DIM)
{
    __shared__ float smem[IN_DIM];        // 128 KB of the 320 KB/WGP pool

    const int row = blockIdx.x;
    const int t   = threadIdx.x;

    // Wave-relative LDS byte offset = low 32 bits of the flat shared address.
    const uint32_t lds_base = (uint32_t)(uintptr_t)(&smem[0]);
    const uint64_t gbase    = (uint64_t)(uintptr_t)(x + (size_t)row * IN_DIM);

    // 32768 floats = 131072 B; 256 lanes * 16 B = 4096 B per pass -> 32 passes.
    constexpr int PASSES = (IN_DIM * 4) / (TPB * 16);
#pragma unroll
    for (int p = 0; p < PASSES; ++p) {
        uint32_t byteoff = (uint32_t)(p * TPB + t) * 16u;
        uint32_t ldsa    = lds_base + byteoff;
        uint64_t ga      = gbase + byteoff;
        asm volatile("global_load_async_to_lds_b128 %0, %1, off th:TH_LOAD_NT"
                     :: "v"(ldsa), "v"(ga) : "memory");
    }
    asm volatile("s_wait_asynccnt 0" ::: "memory");
    __syncthreads();

    const v4u* __restrict__ pk4  = (const v4u*)packed;
    v4f*       __restrict__ out4 = (v4f*)(out + (size_t)row * OUT_DIM);

    // 32768 outputs / (256 threads * 4 per thread) = 32 iterations.
    constexpr int ITERS = OUT_DIM / (TPB * 4);
#pragma unroll 4
    for (int it = 0; it < ITERS; ++it) {
        int j4 = it * TPB + t;            // coalesced u32x4 load / f32x4 store
        v4u pk = pk4[j4];
        v4f r;
        r.x = eval_pair(smem, pk.x);
        r.y = eval_pair(smem, pk.y);
        r.z = eval_pair(smem, pk.z);
        r.w = eval_pair(smem, pk.w);
        __builtin_nontemporal_store(r, &out4[j4]);  // write-once output: keep out of caches
    }
}

// ---------------------------------------------------------------------------
// Launch
// ---------------------------------------------------------------------------
extern "C" void kernel_launch(void* const* d_in, const int* in_sizes, int n_in,
                              void* d_out, int out_size, void* d_ws, size_t ws_size,
                              hipStream_t stream)
{
    const float* x      = (const float*)d_in[0];
    const float* logits = (const float*)d_in[1];
    const int*   idx_a  = (const int*)d_in[2];
    const int*   idx_b  = (const int*)d_in[3];
    float*       out    = (float*)d_out;
    uint32_t*    packed = (uint32_t*)d_ws;   // 128 KB scratch

    pack_idx_kernel<<<OUT_DIM / TPB, TPB, 0, stream>>>(logits, idx_a, idx_b, packed);
    logic_layer_kernel<<<BATCH, TPB, 0, stream>>>(x, packed, out);
}